// UnicycleAccelCurvatureActionSpace_28509992910955
// MI455X (gfx1250) — compile-verified
//
#include <hip/hip_runtime.h>
#include <stdint.h>

// ---------------------------------------------------------------------------
// Batched bandwidth-3 SPD solve for the unicycle speed-fit problem.
//   B = 4096 batches, N = 80 unknowns, M = N+1 = 81 speeds.
// lhs = tridiagonal(A^T A) + constant bandwidth-3 band (DTD) + ridge
//   -> banded Cholesky, O(N*b^2) per batch instead of dense O(N^3/3).
// One thread solves one batch; one wave (32 batches) per block.
// Inputs staged to LDS by the CDNA5 Tensor Data Mover (tensor_load_to_lds),
// outputs staged through LDS and written as coalesced float4 stores.
// ---------------------------------------------------------------------------

#define WAVE 32
#define N_   80
#define M_   81

typedef uint32_t v4u  __attribute__((ext_vector_type(4)));
typedef int32_t  v8i_ __attribute__((ext_vector_type(8)));
typedef int32_t  v4i_ __attribute__((ext_vector_type(4)));

// LDS layout (floats). Padded dxy breaks the stride-160 (== 32 mod 64 banks)
// conflict: padded stride 165 mod 64 = 37 -> conflict-free. After dxy is
// consumed, the same region is reused as the [batch][81] output staging slab.
#define OFF_COS  0                       // [batch][81]  theta -> cos in place
#define OFF_SIN  (OFF_COS + M_ * WAVE)   // [batch][81]
#define OFF_DXY  (OFF_SIN + M_ * WAVE)   // padded: addr(r) = r + (r>>5)
#define SZ_DXY   (2 * N_ * WAVE + ((2 * N_ * WAVE) >> 5))   // 5280 >= 2592
#define OFF_Y    (OFF_DXY + SZ_DXY)      // [p][lane]  rhs, then y
#define OFF_L    (OFF_Y + N_ * WAVE)     // [(p*4+c)][lane]  l1,l2,l3,1/d
#define SM_TOTAL (OFF_L + 4 * N_ * WAVE) // = 23264 floats = 93,056 B

// Build a 1-D TDM descriptor per CDNA5 ISA section 8 and issue the load.
// tensor_dwords limits OOB (reads past it return 0); tile_dwords is the
// transfer size. pad_interval encode 4 == every 32 dwords, pad_amount
// encode 0 == 1 dword.
__device__ __forceinline__ void tdm_load_1d(uint32_t lds_byte, const void* gptr,
                                            uint32_t tensor_dwords,
                                            uint32_t tile_dwords,
                                            uint32_t pad_en, uint32_t pad_interval,
                                            uint32_t pad_amount) {
  uint64_t ga = (uint64_t)(uintptr_t)gptr;
  v4u g0;
  g0[0] = 1u;                                      // count=1, user mode, no gather
  g0[1] = lds_byte;                                // lds_addr
  g0[2] = (uint32_t)ga;                            // global_addr[31:0]
  g0[3] = (uint32_t)((ga >> 32) & 0x01FFFFFFu)     // global_addr[56:32]
        | (2u << 30);                              // type = 2 ("image")
  uint32_t w0  = (2u << 16)                        // data_size = 4 bytes
               | (pad_en << 20) | (pad_interval << 22) | (pad_amount << 25);
  uint32_t td0 = tensor_dwords, td1 = 1u;
  uint32_t tl0 = tile_dwords & 0xFFFFu, tl1 = 1u, tl2 = 0u;
  uint64_t s0  = tensor_dwords;                    // tensor_dim0_stride
  v8i_ g1;
  g1[0] = (int32_t)w0;                             // mask=0, ds=4B, pad cfg
  g1[1] = (int32_t)((td0 & 0xFFFFu) << 16);        // [63:48] = tensor_dim0 lo
  g1[2] = (int32_t)((td0 >> 16) | ((td1 & 0xFFFFu) << 16)); // dim0 hi | dim1 lo
  g1[3] = (int32_t)((td1 >> 16) | (tl0 << 16));    // dim1 hi | tile_dim0
  g1[4] = (int32_t)(tl1 | (tl2 << 16));            // tile_dim1 | tile_dim2
  g1[5] = (int32_t)(uint32_t)(s0 & 0xFFFFFFFFu);   // dim0_stride lo
  g1[6] = (int32_t)(uint32_t)((s0 >> 32) & 0xFFFFu); // dim0_stride hi
  g1[7] = 0;                                       // dim1_stride hi
  v4i_ g2 = {0, 0, 0, 0};                          // tile_dim3 = 0 (unused)
  v4i_ g3 = {0, 0, 0, 0};                          // tile_dim4 = 0 (unused)
#if __has_include(<hip/amd_detail/amd_gfx1250_TDM.h>)
  v8i_ g4 = {0, 0, 0, 0, 0, 0, 0, 0};
  __builtin_amdgcn_tensor_load_to_lds(g0, g1, g2, g3, g4, 0);
#else
  __builtin_amdgcn_tensor_load_to_lds(g0, g1, g2, g3, 0);
#endif
}

__global__ __launch_bounds__(WAVE) void
unicycle_banded_solve_kernel(const float* __restrict__ dxy,
                             const float* __restrict__ theta,
                             const float* __restrict__ v0,
                             float* __restrict__ out, int Btot) {
  __shared__ float sm[SM_TOTAL];
  const int lane = threadIdx.x;                 // wave32: lane == batch-in-block
  const int b0   = blockIdx.x * WAVE;
  const int gb   = b0 + lane;
  const int nbv  = min(WAVE, Btot - b0);

  float* shCos = sm + OFF_COS;
  float* shSin = sm + OFF_SIN;
  float* shDxy = sm + OFF_DXY;
  float* shOut = sm + OFF_DXY;                  // reused after dxy consumed
  float* shY   = sm + OFF_Y;
  float* shL   = sm + OFF_L;

  // ---- Stage inputs with the Tensor Data Mover (async DMA -> LDS) ----
  const uint32_t lbase = (uint32_t)(uintptr_t)sm;   // low 32 bits = LDS offset
  tdm_load_1d(lbase + OFF_COS * 4u, theta + (size_t)b0 * M_,
              (uint32_t)(M_ * nbv), (uint32_t)(M_ * WAVE), 0u, 0u, 0u);
  tdm_load_1d(lbase + OFF_DXY * 4u, dxy + (size_t)b0 * 2 * N_,
              (uint32_t)(2 * N_ * nbv), (uint32_t)(2 * N_ * WAVE),
              1u, 4u, 0u);                          // pad 1 dword / 32 dwords
  __builtin_amdgcn_s_wait_tensorcnt(0);             // single wave: no barrier
  asm volatile("" ::: "memory");

  const float SC = 4.0f;                            // 2/DT, DT = 0.5
  const float G  = 0.0064f;                         // lambda / DT^6
  const float v0v = (gb < Btot) ? v0[gb] : 0.0f;

  // ---- theta -> cos/sin in place (stride-81 LDS rows: conflict-free) ----
  #pragma unroll 1
  for (int j = 0; j < M_; ++j) {
    const int idx = lane * M_ + j;
    const float t = shCos[idx];
    shSin[idx] = __sinf(t);
    shCos[idx] = __cosf(t);
  }

  // ---- rhs[p] = cos(th_{p+1})*bx + sin(th_{p+1})*by, b = (2/DT)*dxy ----
  {
    const int rb = lane * (2 * N_);
    #pragma unroll 1
    for (int p = 0; p < N_; ++p) {
      const int r0 = rb + 2 * p;
      float bx = shDxy[r0 + (r0 >> 5)];
      const int r1 = r0 + 1;
      float by = shDxy[r1 + (r1 >> 5)];
      if (p < N_ - 1) {
        const int r2 = r0 + 2, r3 = r0 + 3;
        bx += shDxy[r2 + (r2 >> 5)];
        by += shDxy[r3 + (r3 >> 5)];
      }
      const int tj = lane * M_ + p + 1;
      shY[p * WAVE + lane] = SC * (shCos[tj] * bx + shSin[tj] * by);
    }
    // v0 corrections: ATA[1,0] = cos(th0 - th1); DTD[:,0] column = G*{-3,3,-1}
    const float c0 = shCos[lane * M_] * shCos[lane * M_ + 1]
                   + shSin[lane * M_] * shSin[lane * M_ + 1];
    shY[0 * WAVE + lane] -= (c0 - 3.0f * G) * v0v;
    shY[1 * WAVE + lane] -= (3.0f * G) * v0v;
    shY[2 * WAVE + lane] += G * v0v;
  }
  asm volatile("" ::: "memory");   // dxy region about to be reused for output

  // ---- Banded Cholesky (half-bandwidth 3) + fused forward substitution ----
  // Rolling state: row p-1 -> (a1,a2,adinv); row p-2 -> (b1,bdinv); row p-3 -> cdinv.
  float a1 = 0.f, a2 = 0.f, adinv = 0.f;
  float b1 = 0.f, bdinv = 0.f, cdinv = 0.f;
  float y1 = 0.f, y2 = 0.f, y3 = 0.f;
  #pragma unroll 1
  for (int p = 0; p < N_; ++p) {
    const int i = p + 1;   // row index in the full (N+1)x(N+1) system
    // D3^T D3 band (stencil [-1,3,-3,1] autocorrelation, boundary-truncated)
    const float d0 = (i == 1) ? 10.f : (i == 2 || i == 78) ? 19.f
                   : (i == 79) ? 10.f : (i == 80) ? 1.f : 20.f;
    const float Add = ((p < N_ - 1) ? 2.0f : 1.0f) + G * d0 + 1e-4f;
    float A1 = 0.f, A2 = 0.f, A3 = 0.f;
    if (p >= 1) {
      const float d1 = (i == 2 || i == 79) ? -12.f : (i == 80) ? -3.f : -15.f;
      const float cp = shCos[lane * M_ + p] * shCos[lane * M_ + p + 1]
                     + shSin[lane * M_ + p] * shSin[lane * M_ + p + 1];
      A1 = cp + G * d1;
    }
    if (p >= 2) A2 = G * ((i == 80) ? 3.f : 6.f);
    if (p >= 3) A3 = -G;

    const float l3   = A3 * cdinv;
    const float l2   = (A2 - l3 * b1) * bdinv;
    const float l1   = (A1 - l3 * a2 - l2 * a1) * adinv;
    const float diag = Add - l3 * l3 - l2 * l2 - l1 * l1;
    const float dinv = __frsqrt_rn(diag);           // 1/L[p][p]
    const float yv   = (shY[p * WAVE + lane] - l3 * y3 - l2 * y2 - l1 * y1) * dinv;
    shY[p * WAVE + lane] = yv;
    shL[(p * 4 + 0) * WAVE + lane] = l1;
    shL[(p * 4 + 1) * WAVE + lane] = l2;
    shL[(p * 4 + 2) * WAVE + lane] = l3;
    shL[(p * 4 + 3) * WAVE + lane] = dinv;
    cdinv = bdinv; b1 = a1; bdinv = adinv;          // roll
    a1 = l1; a2 = l2; adinv = dinv;
    y3 = y2; y2 = y1; y1 = yv;
  }

  // ---- Back substitution; stage x into the [batch][81] output slab ----
  // Staging writes are stride-81 (conflict-free); out[:,0] = v0.
  shOut[lane * M_] = v0v;
  float x1 = 0.f, x2 = 0.f, x3 = 0.f;
  #pragma unroll 1
  for (int p = N_ - 1; p >= 0; --p) {
    float acc = shY[p * WAVE + lane];
    if (p + 1 < N_) acc -= shL[((p + 1) * 4 + 0) * WAVE + lane] * x1;
    if (p + 2 < N_) acc -= shL[((p + 2) * 4 + 1) * WAVE + lane] * x2;
    if (p + 3 < N_) acc -= shL[((p + 3) * 4 + 2) * WAVE + lane] * x3;
    const float xv = acc * shL[(p * 4 + 3) * WAVE + lane];
    shOut[lane * M_ + 1 + p] = xv;
    x3 = x2; x2 = x1; x1 = xv;
  }
  asm volatile("" ::: "memory");   // cross-lane LDS stage -> readback ordering

  // ---- Coalesced output: contiguous 32x81-float slab per full block ----
  if (nbv == WAVE) {
    // 2592 floats = 648 float4 (base is 16B aligned: 32*81*4 % 16 == 0)
    float4* outv = (float4*)(out + (size_t)b0 * M_);
    const float4* smv = (const float4*)shOut;
    #pragma unroll 1
    for (int t = lane; t < (M_ * WAVE) / 4; t += WAVE) {
      outv[t] = smv[t];
    }
  } else {
    // Partial tail block: guarded scalar stores
    #pragma unroll 1
    for (int t = lane; t < M_ * nbv; t += WAVE) {
      out[(size_t)b0 * M_ + t] = shOut[t];
    }
  }
}

extern "C" void kernel_launch(void* const* d_in, const int* in_sizes, int n_in,
                              void* d_out, int out_size, void* d_ws, size_t ws_size,
                              hipStream_t stream) {
  const float* dxy   = (const float*)d_in[0];   // [B, N, 2]
  const float* theta = (const float*)d_in[1];   // [B, N+1]
  const float* v0    = (const float*)d_in[2];   // [B]
  float* out = (float*)d_out;                   // [B, N+1]
  const int B = in_sizes[2];
  const int blocks = (B + WAVE - 1) / WAVE;
  hipLaunchKernelGGL(unicycle_banded_solve_kernel, dim3(blocks), dim3(WAVE), 0,
                     stream, dxy, theta, v0, out, B);
}